// Sampled_Symmetrization_Net_47785806135871
// MI455X (gfx1250) — compile-verified
//
#include <hip/hip_runtime.h>
#include <hip/hip_bf16.h>

typedef __attribute__((ext_vector_type(2))) float v2f;
typedef __attribute__((ext_vector_type(8))) float v8f;

// ---------------------------------------------------------------------------
// Kernel 0: zero the 16 histogram counters in workspace (d_ws is poisoned).
// ---------------------------------------------------------------------------
__global__ void zero_counts_kernel(unsigned int* __restrict__ cnt) {
    if (threadIdx.x < 16) cnt[threadIdx.x] = 0u;
}

// ---------------------------------------------------------------------------
// Kernel 1: histogram of perm values per position (the bandwidth-bound part:
// 16 MB of int32 indices @ 23.3 TB/s ~= 0.7 us floor).
// One int4 (b128) load per perm row; branchless per-thread packed counters
// (4 bins x 16-bit fields in a u64; <=~10 trips/thread -> no overflow).
// Reduce via LDS atomics, then one global atomic per slot per block.
// ---------------------------------------------------------------------------
__global__ void __launch_bounds__(256)
hist_kernel(const int4* __restrict__ p4, int P, unsigned int* __restrict__ cnt) {
    __shared__ unsigned int lcnt[16];
    if (threadIdx.x < 16) lcnt[threadIdx.x] = 0u;
    __syncthreads();

    unsigned long long c0 = 0, c1 = 0, c2 = 0, c3 = 0;
    const int stride = gridDim.x * blockDim.x;
    for (int i = blockIdx.x * blockDim.x + threadIdx.x; i < P; i += stride) {
        int4 a = p4[i];
        c0 += 1ull << ((a.x & 3) << 4);
        c1 += 1ull << ((a.y & 3) << 4);
        c2 += 1ull << ((a.z & 3) << 4);
        c3 += 1ull << ((a.w & 3) << 4);
    }
#pragma unroll
    for (int v = 0; v < 4; ++v) {
        atomicAdd(&lcnt[0 * 4 + v], (unsigned int)((c0 >> (v * 16)) & 0xFFFFull));
        atomicAdd(&lcnt[1 * 4 + v], (unsigned int)((c1 >> (v * 16)) & 0xFFFFull));
        atomicAdd(&lcnt[2 * 4 + v], (unsigned int)((c2 >> (v * 16)) & 0xFFFFull));
        atomicAdd(&lcnt[3 * 4 + v], (unsigned int)((c3 >> (v * 16)) & 0xFFFFull));
    }
    __syncthreads();
    if (threadIdx.x < 16) atomicAdd(&cnt[threadIdx.x], lcnt[threadIdx.x]);
}

// ---------------------------------------------------------------------------
// Kernel 2: one wave32, 15 x V_WMMA_F32_16X16X4_F32.
// All parameters are staged ONCE into pre-padded LDS (padding zeros come from
// a bulk clear), so the WMMA path has no conditionals and no divergent global
// loads: coalesced copy burst -> barrier -> ds_load-fed chained WMMAs.
//   GEMM1: h_pre(16x32) = xpad(16x12) @ W1^Tpad(12x32)   [2 N-tiles x 3 K]
//   GEMM2: y_pre(16x16) = relu(h+b1)  @ W2^Tpad(32x16)   [8 K-chunks]
//   GEMM3: acc(16x16)   = countpad(16x4) @ (y+b2)(4x16)  [1 K-chunk]
// Fragment layouts (ISA 7.12.2, wave32):
//   A:   lane l -> M=l%16, VGPR i -> K = 2*(l/16)+i
//   B:   VGPR r, lane l -> K = r + 2*(l/16), N = l%16
//   C/D: VGPR r, lane l -> M = r + 8*(l/16), N = l%16
// ---------------------------------------------------------------------------
__global__ void __launch_bounds__(32)
mlp_wmma_kernel(const float* __restrict__ x,    // 4x10
                const float* __restrict__ W1,   // 32x10
                const float* __restrict__ b1,   // 32
                const float* __restrict__ W2,   // 4x32
                const float* __restrict__ b2,   // 4
                const unsigned int* __restrict__ cnt, // 16
                float* __restrict__ out) {      // 16 (4x4)
    const int l  = threadIdx.x;      // 0..31
    const int m  = l & 15;
    const int hi = l >> 4;           // 0/1: which half-wave
    const int kb = hi << 1;          // K base (0 or 2) for A/B fragments

    // Flat LDS arena, pre-padded shapes. Total 1968 floats (~7.7 KB).
    __shared__ float smem[1968];
    float* const lds_xp = smem;           // 16x12 = 192   xpad[m][k]
    float* const lds_w1 = smem + 192;     // 12x32 = 384   W1^T[k][n]
    float* const lds_b1 = smem + 576;     // 32
    float* const lds_w2 = smem + 608;     // 32x16 = 512   W2^Tpad[k][n]
    float* const lds_b2 = smem + 1120;    // 16 (padded)
    float* const lds_c  = smem + 1136;    // 16x4  = 64    countpad[m][v]
    float* const lds_h  = smem + 1200;    // 16x32 = 512
    float* const lds_y  = smem + 1712;    // 16x16 = 256

    // ---- bulk clear: provides all zero padding ----
    for (int i = l; i < 1968; i += 32) smem[i] = 0.0f;
    __syncthreads();

    // ---- coalesced staging of parameters (only loads on this path) ----
    for (int i = l; i < 320; i += 32) {       // W1[n][k] -> lds_w1[k][n]
        const int n = i / 10, k = i % 10;
        lds_w1[k * 32 + n] = W1[i];
    }
    for (int i = l; i < 128; i += 32) {       // W2[n][k] -> lds_w2[k][n]
        const int n = i / 32, k = i % 32;
        lds_w2[k * 16 + n] = W2[i];
    }
    for (int i = l; i < 40; i += 32) {        // x[m][k] -> lds_xp[m][k] (stride 12)
        const int mm = i / 10, k = i % 10;
        lds_xp[mm * 12 + k] = x[i];
    }
    lds_b1[l] = b1[l];                        // exactly 32
    if (l < 16) lds_c[l] = (float)cnt[l];     // countpad rows 0..3 (m = l/4)
    if (l < 4)  lds_b2[l] = b2[l];
    __syncthreads();

    // ---- GEMM1: h_pre = xpad @ W1^T, N-tiles t=0,1; K padded 10->12 ----
#pragma unroll
    for (int t = 0; t < 2; ++t) {
        v8f acc = {};
        const int n = t * 16 + m;
#pragma unroll
        for (int kc = 0; kc < 3; ++kc) {
            const int k0 = kc * 4 + kb;
            v2f a, b;
            a[0] = lds_xp[m * 12 + k0];
            a[1] = lds_xp[m * 12 + k0 + 1];
            b[0] = lds_w1[k0 * 32 + n];
            b[1] = lds_w1[(k0 + 1) * 32 + n];
            acc = __builtin_amdgcn_wmma_f32_16x16x4_f32(
                false, a, false, b, (short)0, acc, false, false);
        }
        const float bias = lds_b1[n];
#pragma unroll
        for (int r = 0; r < 8; ++r)
            lds_h[(r + (hi << 3)) * 32 + n] = fmaxf(acc[r] + bias, 0.0f);
    }
    __syncthreads();

    // ---- GEMM2: y_pre = h(16x32) @ W2^Tpad(32x16), 8 K-chunks ----
    v8f acc2 = {};
#pragma unroll
    for (int kc = 0; kc < 8; ++kc) {
        const int k0 = kc * 4 + kb;
        v2f a, b;
        a[0] = lds_h[m * 32 + k0];
        a[1] = lds_h[m * 32 + k0 + 1];
        b[0] = lds_w2[k0 * 16 + m];
        b[1] = lds_w2[(k0 + 1) * 16 + m];
        acc2 = __builtin_amdgcn_wmma_f32_16x16x4_f32(
            false, a, false, b, (short)0, acc2, false, false);
    }
    const float bias2 = lds_b2[m];            // zero for padded columns
#pragma unroll
    for (int r = 0; r < 8; ++r)
        lds_y[(r + (hi << 3)) * 16 + m] = acc2[r] + bias2;  // rows 0..3 clean
    __syncthreads();

    // ---- GEMM3: acc = countpad(16x4) @ y(4x16); K=4 exactly, one WMMA ----
    v2f a3, b3;
    a3[0] = lds_c[m * 4 + kb];
    a3[1] = lds_c[m * 4 + kb + 1];
    b3[0] = lds_y[(kb + 0) * 16 + m];
    b3[1] = lds_y[(kb + 1) * 16 + m];
    v8f c3 = {};
    c3 = __builtin_amdgcn_wmma_f32_16x16x4_f32(
        false, a3, false, b3, (short)0, c3, false, false);

    // D: lane l<16 -> M=r, N=l.  out[n*4+o] = acc[n][o] with n=r<4, o=l<4.
    if (l < 4) {
#pragma unroll
        for (int r = 0; r < 4; ++r) out[r * 4 + l] = c3[r];
    }
}

// ---------------------------------------------------------------------------
extern "C" void kernel_launch(void* const* d_in, const int* in_sizes, int n_in,
                              void* d_out, int out_size, void* d_ws, size_t ws_size,
                              hipStream_t stream) {
    const float* x     = (const float*)d_in[0];
    const int*   perms = (const int*)d_in[1];   // (P,4) int32 per harness convention
    const float* W1    = (const float*)d_in[2];
    const float* b1    = (const float*)d_in[3];
    const float* W2    = (const float*)d_in[4];
    const float* b2    = (const float*)d_in[5];
    float*       out   = (float*)d_out;
    unsigned int* cnt  = (unsigned int*)d_ws;   // 16 counters

    const int P = in_sizes[1] / 4;

    zero_counts_kernel<<<1, 32, 0, stream>>>(cnt);

    // ~8 rows per thread; bandwidth-bound sweep of the 16 MB index stream.
    int blocks = (P + 2047) / 2048;
    if (blocks < 1) blocks = 1;
    if (blocks > 1024) blocks = 1024;
    hist_kernel<<<blocks, 256, 0, stream>>>((const int4*)perms, P, cnt);

    mlp_wmma_kernel<<<1, 32, 0, stream>>>(x, W1, b1, W2, b2, cnt, out);
}